// MPL_BG_61323543053001
// MI455X (gfx1250) — compile-verified
//
#include <hip/hip_runtime.h>
#include <hip/hip_bf16.h>

// ---------------------------------------------------------------------------
// GNN layer: fc1 (WMMA f32 GEMM) -> edge SpMM (atomic f32) -> BN stats ->
// fused BN+tanh -> fc2 over concat (two-phase WMMA f32 GEMM) -> tanh.
// fp32 end-to-end, using V_WMMA_F32_16X16X4_F32 for both dense GEMMs.
// ---------------------------------------------------------------------------

typedef float v2f __attribute__((ext_vector_type(2)));
typedef float v8f __attribute__((ext_vector_type(8)));

#define FT 128

__device__ __forceinline__ v8f wmma16x16x4(v2f a, v2f b, v8f c) {
    // 8 args: (neg_a, A, neg_b, B, c_mod, C, reuse_a, reuse_b)
    return __builtin_amdgcn_wmma_f32_16x16x4_f32(false, a, false, b, (short)0, c,
                                                 false, false);
}

// ---------------------------------------------------------------------------
// GEMM1: seq_fts[i][o] = sum_k seq[i][k] * W1[o][k]   (N x 128 @ 128 x 128)
// Block: 256 threads = 8 waves; tile 64 rows x 128 cols.
// Wave (wm, wn): wm in 0..3 -> 16-row strip, wn in 0..1 -> 64-col strip.
// ---------------------------------------------------------------------------
__global__ __launch_bounds__(256) void gemm_fc1_kernel(
    const float* __restrict__ seq, const float* __restrict__ W1,
    float* __restrict__ seq_fts, int nrows) {
    __shared__ float sW[FT * FT];  // W1 row-major [o][k], 64 KB

    const int tid = threadIdx.x;
    {
        const float4* src = (const float4*)W1;
        float4* dst = (float4*)sW;
        #pragma unroll
        for (int i = 0; i < 16; ++i) dst[tid + i * 256] = src[tid + i * 256];
    }
    __syncthreads();

    const int wid = tid >> 5, lane = tid & 31;
    const int l = lane & 15, h = lane >> 4;
    const int m0 = blockIdx.x * 64 + (wid >> 1) * 16;
    const int n0 = (wid & 1) * 64;
    const int m = min(m0 + l, nrows - 1);  // clamp; stores are guarded
    const float* arow = seq + (size_t)m * FT;

    v8f acc0 = {}, acc1 = {}, acc2 = {}, acc3 = {};
    #pragma unroll 4
    for (int k0 = 0; k0 < FT; k0 += 4) {
        const int k = k0 + 2 * h;  // half of wave takes K={0,1}, other K={2,3}
        v2f a = *(const v2f*)(arow + k);
        v2f b0 = *(const v2f*)(sW + (size_t)(n0 + 0  + l) * FT + k);
        v2f b1 = *(const v2f*)(sW + (size_t)(n0 + 16 + l) * FT + k);
        v2f b2 = *(const v2f*)(sW + (size_t)(n0 + 32 + l) * FT + k);
        v2f b3 = *(const v2f*)(sW + (size_t)(n0 + 48 + l) * FT + k);
        acc0 = wmma16x16x4(a, b0, acc0);
        acc1 = wmma16x16x4(a, b1, acc1);
        acc2 = wmma16x16x4(a, b2, acc2);
        acc3 = wmma16x16x4(a, b3, acc3);
    }

    #pragma unroll
    for (int r = 0; r < 8; ++r) {
        const int row = m0 + r + 8 * h;  // D layout: VGPR r -> row r + 8*half
        if (row < nrows) {
            float* orow = seq_fts + (size_t)row * FT + n0 + l;
            orow[0]  = acc0[r];
            orow[16] = acc1[r];
            orow[32] = acc2[r];
            orow[48] = acc3[r];
        }
    }
}

// ---------------------------------------------------------------------------
// SpMM: agg[row[e]][:] += val[e] * seq_fts[col[e]][:].  One wave per edge.
// seq_fts (51 MB) and agg live in the 192 MB L2 -> atomics stay on-chip.
// ---------------------------------------------------------------------------
__global__ __launch_bounds__(256) void spmm_kernel(
    const int* __restrict__ adj_row, const int* __restrict__ adj_col,
    const float* __restrict__ adj_val, const float* __restrict__ seq_fts,
    float* __restrict__ agg, int nedges) {
    const int wid = threadIdx.x >> 5, lane = threadIdx.x & 31;
    const int e = blockIdx.x * 8 + wid;
    if (e >= nedges) return;
    const int r = adj_row[e], c = adj_col[e];
    const float v = adj_val[e];
    const float4 x = ((const float4*)(seq_fts + (size_t)c * FT))[lane];
    float* dst = agg + (size_t)r * FT + lane * 4;
    unsafeAtomicAdd(dst + 0, v * x.x);
    unsafeAtomicAdd(dst + 1, v * x.y);
    unsafeAtomicAdd(dst + 2, v * x.z);
    unsafeAtomicAdd(dst + 3, v * x.w);
}

// ---------------------------------------------------------------------------
// BN statistics: per-feature sum / sum-of-squares, coalesced row sweeps,
// folded into global accumulators with f32 atomics.
// ---------------------------------------------------------------------------
__global__ __launch_bounds__(256) void bn_partial_kernel(
    const float* __restrict__ agg, float* __restrict__ sum,
    float* __restrict__ sq, int nrows) {
    const int f = threadIdx.x & 127;
    const int rh = threadIdx.x >> 7;           // 2 rows in flight per block
    const int rend = min(nrows, (int)(blockIdx.x + 1) * 512);
    float s = 0.f, q = 0.f;
    for (int r = blockIdx.x * 512 + rh; r < rend; r += 2) {
        const float x = agg[(size_t)r * FT + f];
        s += x;
        q += x * x;
    }
    unsafeAtomicAdd(sum + f, s);
    unsafeAtomicAdd(sq + f, q);
}

__global__ void bn_finalize_kernel(const float* __restrict__ sum,
                                   const float* __restrict__ sq,
                                   const float* __restrict__ gamma,
                                   const float* __restrict__ beta,
                                   float* __restrict__ scale,
                                   float* __restrict__ shift, int nrows) {
    const int f = threadIdx.x;
    const float invN = 1.0f / (float)nrows;
    const float mean = sum[f] * invN;
    const float var = sq[f] * invN - mean * mean;  // biased variance
    const float inv = rsqrtf(var + 1e-5f);
    const float sc = gamma[f] * inv;
    scale[f] = sc;
    shift[f] = beta[f] - mean * sc;
}

// hbn = tanh(agg * scale + shift), written over the dead seq_fts buffer.
__global__ __launch_bounds__(256) void bn_tanh_kernel(
    const float* __restrict__ agg, const float* __restrict__ scale,
    const float* __restrict__ shift, float* __restrict__ hbn, int n) {
    const int i = blockIdx.x * 256 + threadIdx.x;
    if (i < n) {
        const int f = i & (FT - 1);
        hbn[i] = tanhf(agg[i] * scale[f] + shift[f]);
    }
}

// ---------------------------------------------------------------------------
// GEMM2: out = tanh([seq_self | hbn] @ W2^T).  K=256 as two 128-K phases;
// the matching 64 KB half of W2 is restaged in LDS per phase.
// ---------------------------------------------------------------------------
__global__ __launch_bounds__(256) void gemm_fc2_kernel(
    const float* __restrict__ seq_self, const float* __restrict__ hbn,
    const float* __restrict__ W2, float* __restrict__ out, int nrows) {
    __shared__ float sW[FT * FT];  // one K-half of W2, [o][k], 64 KB

    const int tid = threadIdx.x;
    const int wid = tid >> 5, lane = tid & 31;
    const int l = lane & 15, h = lane >> 4;
    const int m0 = blockIdx.x * 64 + (wid >> 1) * 16;
    const int n0 = (wid & 1) * 64;
    const int m = min(m0 + l, nrows - 1);

    v8f acc0 = {}, acc1 = {}, acc2 = {}, acc3 = {};
    const float4* w4 = (const float4*)W2;

    for (int phase = 0; phase < 2; ++phase) {
        __syncthreads();  // previous phase done reading sW
        for (int i = tid; i < 4096; i += 256) {
            const int o = i >> 5, kq = i & 31;       // W2 row o, float4 kq
            ((float4*)sW)[i] = w4[o * 64 + phase * 32 + kq];
        }
        __syncthreads();

        const float* arow =
            (phase == 0 ? seq_self : hbn) + (size_t)m * FT;
        #pragma unroll 4
        for (int k0 = 0; k0 < FT; k0 += 4) {
            const int k = k0 + 2 * h;
            v2f a = *(const v2f*)(arow + k);
            v2f b0 = *(const v2f*)(sW + (size_t)(n0 + 0  + l) * FT + k);
            v2f b1 = *(const v2f*)(sW + (size_t)(n0 + 16 + l) * FT + k);
            v2f b2 = *(const v2f*)(sW + (size_t)(n0 + 32 + l) * FT + k);
            v2f b3 = *(const v2f*)(sW + (size_t)(n0 + 48 + l) * FT + k);
            acc0 = wmma16x16x4(a, b0, acc0);
            acc1 = wmma16x16x4(a, b1, acc1);
            acc2 = wmma16x16x4(a, b2, acc2);
            acc3 = wmma16x16x4(a, b3, acc3);
        }
    }

    #pragma unroll
    for (int r = 0; r < 8; ++r) {
        const int row = m0 + r + 8 * h;
        if (row < nrows) {
            float* orow = out + (size_t)row * FT + n0 + l;
            orow[0]  = tanhf(acc0[r]);
            orow[16] = tanhf(acc1[r]);
            orow[32] = tanhf(acc2[r]);
            orow[48] = tanhf(acc3[r]);
        }
    }
}

// ---------------------------------------------------------------------------
// Launch
// ---------------------------------------------------------------------------
extern "C" void kernel_launch(void* const* d_in, const int* in_sizes, int n_in,
                              void* d_out, int out_size, void* d_ws, size_t ws_size,
                              hipStream_t stream) {
    const float* seq_self = (const float*)d_in[0];
    const float* seq      = (const float*)d_in[1];
    const int*   adj_row  = (const int*)d_in[2];
    const int*   adj_col  = (const int*)d_in[3];
    const float* adj_val  = (const float*)d_in[4];
    const float* W1       = (const float*)d_in[5];
    const float* W2       = (const float*)d_in[6];
    const float* gamma    = (const float*)d_in[7];
    const float* beta     = (const float*)d_in[8];
    float* out = (float*)d_out;

    const int nrows  = in_sizes[1] / FT;  // N = 100000
    const int nedges = in_sizes[2];       // E = 3200000

    float* seq_fts = (float*)d_ws;                       // [N,128], reused as hbn
    float* agg     = seq_fts + (size_t)nrows * FT;       // [N,128]
    float* stats   = agg + (size_t)nrows * FT;           // sum|sq|scale|shift
    float* s_sum = stats, *s_sq = stats + FT;
    float* s_scale = stats + 2 * FT, *s_shift = stats + 3 * FT;

    hipMemsetAsync(agg, 0, (size_t)nrows * FT * sizeof(float), stream);
    hipMemsetAsync(stats, 0, 2 * FT * sizeof(float), stream);

    const int mblocks = (nrows + 63) / 64;
    gemm_fc1_kernel<<<mblocks, 256, 0, stream>>>(seq, W1, seq_fts, nrows);
    spmm_kernel<<<(nedges + 7) / 8, 256, 0, stream>>>(adj_row, adj_col, adj_val,
                                                      seq_fts, agg, nedges);
    bn_partial_kernel<<<(nrows + 511) / 512, 256, 0, stream>>>(agg, s_sum, s_sq,
                                                               nrows);
    bn_finalize_kernel<<<1, FT, 0, stream>>>(s_sum, s_sq, gamma, beta, s_scale,
                                             s_shift, nrows);
    bn_tanh_kernel<<<(nrows * FT + 255) / 256, 256, 0, stream>>>(
        agg, s_scale, s_shift, seq_fts, nrows * FT);
    gemm_fc2_kernel<<<mblocks, 256, 0, stream>>>(seq_self, seq_fts, W2, out,
                                                 nrows);
}